// GCN_layer_68049461838442
// MI455X (gfx1250) — compile-verified
//
#include <hip/hip_runtime.h>

#define D 256  // D_IN == D_OUT == 256

typedef float v2f __attribute__((ext_vector_type(2)));
typedef float v8f __attribute__((ext_vector_type(8)));

// ---------------------------------------------------------------------------
// 1) Zero the output (harness poisons d_out; must re-zero deterministically).
// ---------------------------------------------------------------------------
__global__ void gcn_zero_out(float4* __restrict__ out, int n4) {
    int i = blockIdx.x * blockDim.x + threadIdx.x;
    if (i < n4) out[i] = make_float4(0.f, 0.f, 0.f, 0.f);
}

// ---------------------------------------------------------------------------
// 2) deg[i] = 1 (self-loop), then count src occurrences with int atomics.
// ---------------------------------------------------------------------------
__global__ void gcn_init_deg(int* __restrict__ deg, int n) {
    int i = blockIdx.x * blockDim.x + threadIdx.x;
    if (i < n) deg[i] = 1;
}

__global__ void gcn_count_deg(const int* __restrict__ src, int* __restrict__ deg, int e) {
    int i = blockIdx.x * blockDim.x + threadIdx.x;
    if (i < e) atomicAdd(&deg[src[i]], 1);
}

// ---------------------------------------------------------------------------
// 3) w[i] = dinv[i]^2 = 1/deg[i]   (deg >= 1 always, so no div-by-zero)
// ---------------------------------------------------------------------------
__global__ void gcn_weights(const int* __restrict__ deg, float* __restrict__ wgt, int n) {
    int i = blockIdx.x * blockDim.x + threadIdx.x;
    if (i < n) wgt[i] = 1.0f / (float)deg[i];
}

// ---------------------------------------------------------------------------
// 4) XW = X @ W with V_WMMA_F32_16X16X4_F32 (full fp32, faithful to reference)
//    Block = 256 threads = 8 waves. Wave (w>>2) picks row-tile within the
//    block's 32-row slab; (w&3) picks a 64-wide N strip (4 x 16x16 tiles).
//    A lane layout (16x4 f32): lane L holds row (L&15), K pair (L>>4)*2 —
//    contiguous in memory -> single b64 load per k-step.
// ---------------------------------------------------------------------------
__global__ void gcn_gemm_xw(const float* __restrict__ X,
                            const float* __restrict__ W,
                            float* __restrict__ XW) {
    const int lane = threadIdx.x & 31;
    const int wave = threadIdx.x >> 5;
    const int row0 = (blockIdx.x * 2 + (wave >> 2)) * 16;
    const int col0 = (wave & 3) * 64;
    const int m    = lane & 15;          // row within A tile / col within B tile
    const int kh   = (lane >> 4) * 2;    // K sub-offset: lanes 0-15 -> K0..1, 16-31 -> K2..3

    v8f c0 = {}, c1 = {}, c2 = {}, c3 = {};

    const float* arow = X + (size_t)(row0 + m) * D;
    for (int k0 = 0; k0 < D; k0 += 4) {
        v2f a = *(const v2f*)(arow + k0 + kh);               // global_load_b64
        const float* wr0 = W + (size_t)(k0 + kh) * D + col0 + m;
        const float* wr1 = wr0 + D;                          // next K row of W
        v2f b0 = { wr0[0],  wr1[0]  };
        v2f b1 = { wr0[16], wr1[16] };
        v2f b2 = { wr0[32], wr1[32] };
        v2f b3 = { wr0[48], wr1[48] };
        c0 = __builtin_amdgcn_wmma_f32_16x16x4_f32(false, a, false, b0, (short)0, c0, false, false);
        c1 = __builtin_amdgcn_wmma_f32_16x16x4_f32(false, a, false, b1, (short)0, c1, false, false);
        c2 = __builtin_amdgcn_wmma_f32_16x16x4_f32(false, a, false, b2, (short)0, c2, false, false);
        c3 = __builtin_amdgcn_wmma_f32_16x16x4_f32(false, a, false, b3, (short)0, c3, false, false);
    }

    // C/D layout: VGPR v -> row (v + 8*(lane>>4)), col (lane&15)
    const int mrow = 8 * (lane >> 4);
    float* o = XW + (size_t)row0 * D + col0 + m;
    #pragma unroll
    for (int v = 0; v < 8; ++v) {
        float* orow = o + (size_t)(v + mrow) * D;
        orow[0]  = c0[v];
        orow[16] = c1[v];
        orow[32] = c2[v];
        orow[48] = c3[v];
    }
}

// ---------------------------------------------------------------------------
// 5) Edge scatter: one wave per edge.
//    out[row] += w[col] * XW[col]   (256 floats = 8 per lane, float4 gathers,
//    global_atomic_add_f32 scatters; XW and out both live in the 192MB L2).
// ---------------------------------------------------------------------------
__global__ void gcn_scatter(const int* __restrict__ eidx,
                            const float* __restrict__ wgt,
                            const float* __restrict__ XW,
                            float* __restrict__ out, int E) {
    const int gw   = (int)((blockIdx.x * (size_t)blockDim.x + threadIdx.x) >> 5);
    const int lane = threadIdx.x & 31;
    if (gw >= E) return;

    const int r = eidx[gw];        // row  (destination of the scatter-add)
    const int c = eidx[E + gw];    // col  (source row, provides the weight)
    const float wv = wgt[c];

    const float4* src = (const float4*)(XW + (size_t)c * D) + lane;  // b128 gather
    float*        dst = out + (size_t)r * D + lane * 4;

    float4 x0 = src[0];
    float4 x1 = src[32];           // +128 floats
    unsafeAtomicAdd(dst + 0, wv * x0.x);
    unsafeAtomicAdd(dst + 1, wv * x0.y);
    unsafeAtomicAdd(dst + 2, wv * x0.z);
    unsafeAtomicAdd(dst + 3, wv * x0.w);
    float* dst1 = dst + 128;
    unsafeAtomicAdd(dst1 + 0, wv * x1.x);
    unsafeAtomicAdd(dst1 + 1, wv * x1.y);
    unsafeAtomicAdd(dst1 + 2, wv * x1.z);
    unsafeAtomicAdd(dst1 + 3, wv * x1.w);
}

// ---------------------------------------------------------------------------
// 6) Self-loop: out[i] += (1/deg[i]) * XW[i]. Runs after the atomic scatter
//    (stream-ordered), each element owned by one thread -> plain RMW.
// ---------------------------------------------------------------------------
__global__ void gcn_selfloop(const float* __restrict__ wgt,
                             const float4* __restrict__ XW4,
                             float4* __restrict__ out4, int n4) {
    int i = blockIdx.x * blockDim.x + threadIdx.x;
    if (i >= n4) return;
    const float wv = wgt[i >> 6];          // 64 float4 per 256-wide row
    float4 x = XW4[i];
    float4 o = out4[i];
    o.x += wv * x.x; o.y += wv * x.y; o.z += wv * x.z; o.w += wv * x.w;
    out4[i] = o;
}

// ---------------------------------------------------------------------------
extern "C" void kernel_launch(void* const* d_in, const int* in_sizes, int n_in,
                              void* d_out, int out_size, void* d_ws, size_t ws_size,
                              hipStream_t stream) {
    const float* X    = (const float*)d_in[0];   // [N, 256]
    const int*   eidx = (const int*)d_in[1];     // [2, E] (row-major: src then dst)
    const float* W    = (const float*)d_in[2];   // [256, 256]
    float*       out  = (float*)d_out;           // [N, 256]

    const int n = in_sizes[0] / D;               // 100000
    const int E = in_sizes[1] / 2;               // 3200000

    // Workspace layout: deg (int[n]) | wgt (float[n]) | XW (float[n*256])
    char* ws = (char*)d_ws;
    size_t off = 0;
    int* deg = (int*)(ws + off);
    off = (off + (size_t)n * sizeof(int)   + 255) & ~(size_t)255;
    float* wgt = (float*)(ws + off);
    off = (off + (size_t)n * sizeof(float) + 255) & ~(size_t)255;
    float* XW = (float*)(ws + off);

    const int n4 = n * (D / 4);                  // float4 count of out / XW

    gcn_zero_out<<<(n4 + 255) / 256, 256, 0, stream>>>((float4*)out, n4);
    gcn_init_deg<<<(n + 255) / 256, 256, 0, stream>>>(deg, n);
    gcn_count_deg<<<(E + 255) / 256, 256, 0, stream>>>(eidx, deg, E);
    gcn_weights<<<(n + 255) / 256, 256, 0, stream>>>(deg, wgt, n);

    // 32 output rows per block; 100000 = 3125 * 32 exactly.
    gcn_gemm_xw<<<n / 32, 256, 0, stream>>>(X, W, XW);

    // one wave (32 lanes) per edge; 8 waves per 256-thread block
    gcn_scatter<<<(E + 7) / 8, 256, 0, stream>>>(eidx, wgt, XW, out, E);

    gcn_selfloop<<<(n4 + 255) / 256, 256, 0, stream>>>(wgt, (const float4*)XW,
                                                       (float4*)out, n4);
}